// Uncert_control_MODNET_13572096655540
// MI455X (gfx1250) — compile-verified
//
#include <hip/hip_runtime.h>

// Elementwise map, memory-bound (12 B/elem, 240 MB stream @ 23.3 TB/s ~ 10 us).
// All trig arguments are multiples of 2*pi, so we use raw v_sin_f32/v_cos_f32
// (which compute sin/cos(2*pi*arg)) with pre-scaled args -> 1 TRANS op each.
// Non-temporal loads/stores (CDNA5 TH=NT) since the stream exceeds the 192 MB L2.

// Native clang vector types (HIP float4/float2 are structs and are rejected by
// the nontemporal builtins).
typedef __attribute__((ext_vector_type(4))) float v4f;
typedef __attribute__((ext_vector_type(2))) float v2f;

__device__ __forceinline__ float compute_one(float x, float y) {
    // masks
    const bool m1     = (y >= 1.0f) && (y < 1.5f);
    const bool active = (y >= 1.0f) && (y < 2.0f);
    const float y_safe = active ? y : 1.0f;

    // sin(4*pi*x*y) == v_sin(2*x*y)
    const float s1 = __builtin_amdgcn_sinf(2.0f * x * y);

    // inner = 2x - 1 + x*(x-1)*(2 + s1)/y_safe   (y_safe in [1,2) when it matters)
    const float inv_y = __builtin_amdgcn_rcpf(y_safe);
    const float inner = 2.0f * x - 1.0f + x * (x - 1.0f) * (2.0f + s1) * inv_y;

    // trig = m1 ? cos(pi*y) : sin(2*pi*y)  -> v_cos(0.5y) / v_sin(y)
    const float c  = __builtin_amdgcn_cosf(0.5f * y);
    const float s2 = __builtin_amdgcn_sinf(y);
    const float trig = m1 ? c : s2;

    const float poly = y * y - 3.0f * y + 2.0f;
    const float val  = poly * (x * x - x + inner * (x + trig));
    return active ? val : 0.0f;
}

__global__ __launch_bounds__(256) void uncert_map_kernel(
    const float* __restrict__ in,   // (n, 2) interleaved x,y
    float* __restrict__ out,        // (n)
    int n)
{
    const int t  = blockIdx.x * blockDim.x + threadIdx.x;
    const int e0 = t * 2;  // first element index handled by this thread

    if (e0 + 1 < n) {
        // 128-bit coalesced non-temporal load: {x0, y0, x1, y1}
        const v4f v = __builtin_nontemporal_load(
            reinterpret_cast<const v4f*>(in + (size_t)e0 * 2));
        v2f r;
        r.x = compute_one(v.x, v.y);
        r.y = compute_one(v.z, v.w);
        __builtin_nontemporal_store(r, reinterpret_cast<v2f*>(out + e0));
    } else if (e0 < n) {
        // odd tail (not hit for N = 20,000,000, but keep it correct)
        const float x = in[(size_t)e0 * 2 + 0];
        const float y = in[(size_t)e0 * 2 + 1];
        out[e0] = compute_one(x, y);
    }
}

extern "C" void kernel_launch(void* const* d_in, const int* in_sizes, int n_in,
                              void* d_out, int out_size, void* d_ws, size_t ws_size,
                              hipStream_t stream) {
    (void)n_in; (void)d_ws; (void)ws_size; (void)in_sizes;
    const float* in = (const float*)d_in[0];  // (N,2) float32
    // d_in[1] is the scalar 'a' (unused by the reference math)
    float* out = (float*)d_out;
    const int n = out_size;                    // N rows

    const int pairs  = (n + 1) / 2;            // 2 elements per thread
    const int block  = 256;                    // 8 wave32s per block
    const int grid   = (pairs + block - 1) / block;
    uncert_map_kernel<<<grid, block, 0, stream>>>(in, out, n);
}